// SparseSEWResNet_69965017252012
// MI455X (gfx1250) — compile-verified
//
#include <hip/hip_runtime.h>
#include <hip/hip_bf16.h>

typedef float v2f __attribute__((ext_vector_type(2)));
typedef float v8f __attribute__((ext_vector_type(8)));

#define C 128
#define STRIDE_B 4194304   // 16*512*512 = 2^22
#define STRIDE_T 262144    // 512*512   = 2^18
#define STRIDE_H 512       // 2^9
#define SENTINEL 33554432  // 8*STRIDE_B = 2^25

// ---------------------------------------------------------------------------
// Pack W[128][128] into WMMA fragment order:
//   for K-step s (0..31), col-tile t (0..7), lane L (0..31):
//     pair = ( W[4s + 2*(L>>4)    ][t*16 + (L&15)],
//              W[4s + 2*(L>>4) + 1][t*16 + (L&15)] )
// stored contiguously -> one global_load_b64 per B fragment, fully coalesced.
// ---------------------------------------------------------------------------
__global__ void pack_w(const float* __restrict__ W, float* __restrict__ Wp)
{
    int i = blockIdx.x * blockDim.x + threadIdx.x;   // 32*8*32 = 8192 fragments
    if (i < 32 * 8 * 32) {
        int L = i & 31;
        int t = (i >> 5) & 7;
        int s = i >> 8;
        int koff = (L >> 4) * 2;
        int n = t * 16 + (L & 15);
        Wp[i * 2 + 0] = W[(s * 4 + koff) * C + n];
        Wp[i * 2 + 1] = W[(s * 4 + koff + 1) * C + n];
    }
}

// ---------------------------------------------------------------------------
// Fused: out2 = LN2( relu(LN1(feats@W1)) @ W2 ), plus column-sum atomics.
// One wave = one 16-row tile; full 128-wide output kept in 8 f32 accumulators.
// ---------------------------------------------------------------------------
__global__ __launch_bounds__(64) void fused_mlp(
    const float* __restrict__ feats,
    const float* __restrict__ W1p, const float* __restrict__ g1, const float* __restrict__ b1,
    const float* __restrict__ W2p, const float* __restrict__ g2, const float* __restrict__ b2,
    float* __restrict__ X2, float* __restrict__ colsum, int N)
{
    __shared__ float tile[2][16 * 132];   // per-wave 16x128 staging (padded pitch)

    const int wave = threadIdx.x >> 5;
    const int lane = threadIdx.x & 31;
    const int tileIdx = blockIdx.x * 2 + wave;
    const int row0 = tileIdx * 16;
    const int m    = lane & 15;     // N-col within 16 / M-row for A frags
    const int half = lane >> 4;     // 0: K0/K1, rows 0-7 ; 1: K2/K3, rows 8-15
    const int koff = half * 2;

    const v2f* W1f = (const v2f*)W1p;
    const v2f* W2f = (const v2f*)W2p;

    v8f acc[8];
    #pragma unroll
    for (int t = 0; t < 8; ++t) acc[t] = {};

    // ---- GEMM1: A = feats tile (16x4), B = packed W1 fragments ----
    {
        int rA = row0 + m; if (rA >= N) rA = N - 1;
        const float* arow = feats + (size_t)rA * C;
        for (int s = 0; s < 32; ++s) {
            v2f a = *(const v2f*)(arow + s * 4 + koff);
            #pragma unroll
            for (int t = 0; t < 8; ++t) {
                v2f bf = W1f[(s * 8 + t) * 32 + lane];
                acc[t] = __builtin_amdgcn_wmma_f32_16x16x4_f32(
                    false, a, false, bf, (short)0, acc[t], false, false);
            }
        }
    }

    // ---- LN1 + ReLU -> LDS tile ----
    float* myTile = &tile[wave][0];
    #pragma unroll
    for (int v = 0; v < 8; ++v) {
        float s = 0.f, sq = 0.f;
        #pragma unroll
        for (int t = 0; t < 8; ++t) { float x = acc[t][v]; s += x; sq += x * x; }
        #pragma unroll
        for (int off = 1; off < 16; off <<= 1) {
            s  += __shfl_xor(s,  off, 32);
            sq += __shfl_xor(sq, off, 32);
        }
        float mu  = s * (1.f / 128.f);
        float var = sq * (1.f / 128.f) - mu * mu;
        float rs  = rsqrtf(var + 1e-5f);
        int rowm = v + half * 8;
        #pragma unroll
        for (int t = 0; t < 8; ++t) {
            int n = t * 16 + m;
            float y = (acc[t][v] - mu) * rs * g1[n] + b1[n];
            myTile[rowm * 132 + n] = fmaxf(y, 0.f);
        }
    }
    __syncthreads();

    // ---- GEMM2: A from LDS tile (b64 loads), B = packed W2 fragments ----
    #pragma unroll
    for (int t = 0; t < 8; ++t) acc[t] = {};
    for (int s = 0; s < 32; ++s) {
        v2f a = *(const v2f*)(myTile + m * 132 + s * 4 + koff);  // 8B-aligned
        #pragma unroll
        for (int t = 0; t < 8; ++t) {
            v2f bf = W2f[(s * 8 + t) * 32 + lane];
            acc[t] = __builtin_amdgcn_wmma_f32_16x16x4_f32(
                false, a, false, bf, (short)0, acc[t], false, false);
        }
    }

    // ---- LN2 -> X2, column partial sums ----
    float cs[8];
    #pragma unroll
    for (int t = 0; t < 8; ++t) cs[t] = 0.f;

    #pragma unroll
    for (int v = 0; v < 8; ++v) {
        float s = 0.f, sq = 0.f;
        #pragma unroll
        for (int t = 0; t < 8; ++t) { float x = acc[t][v]; s += x; sq += x * x; }
        #pragma unroll
        for (int off = 1; off < 16; off <<= 1) {
            s  += __shfl_xor(s,  off, 32);
            sq += __shfl_xor(sq, off, 32);
        }
        float mu  = s * (1.f / 128.f);
        float var = sq * (1.f / 128.f) - mu * mu;
        float rs  = rsqrtf(var + 1e-5f);
        int rowm = v + half * 8;
        int grow = row0 + rowm;
        #pragma unroll
        for (int t = 0; t < 8; ++t) {
            int n = t * 16 + m;
            float y = (acc[t][v] - mu) * rs * g2[n] + b2[n];
            if (grow < N) X2[(size_t)grow * C + n] = y;
            cs[t] += (grow < N) ? y : 0.f;
        }
    }
    #pragma unroll
    for (int t = 0; t < 8; ++t) {
        float c = cs[t];
        c += __shfl_xor(c, 16, 32);
        if (half == 0) atomicAdd(&colsum[t * 16 + m], c);
    }
}

// ---------------------------------------------------------------------------
// SE gate: scale = sigmoid(relu(mean @ se_w1 + b1) @ se_w2 + b2)
// ---------------------------------------------------------------------------
__global__ __launch_bounds__(128) void se_kernel(
    const float* __restrict__ colsum,
    const float* __restrict__ w1, const float* __restrict__ b1,
    const float* __restrict__ w2, const float* __restrict__ b2,
    float* __restrict__ scale, int N)
{
    __shared__ float sq[128];
    __shared__ float h[32];
    int t = threadIdx.x;
    sq[t] = colsum[t] / (float)N;
    __syncthreads();
    if (t < 32) {
        float a = b1[t];
        for (int c = 0; c < 128; ++c) a += sq[c] * w1[c * 32 + t];
        h[t] = fmaxf(a, 0.f);
    }
    __syncthreads();
    float a = b2[t];
    for (int j = 0; j < 32; ++j) a += h[j] * w2[j * 128 + t];
    scale[t] = 1.f / (1.f + __expf(-a));
}

// ---------------------------------------------------------------------------
// Sort / unique machinery (7-pass 4-bit LSD radix, stable)
// ---------------------------------------------------------------------------
#define RITEMS 16
#define RELEMS 4096   // 256 threads * 16 items

__global__ void zero_f(float* p, int n) {
    int i = blockIdx.x * blockDim.x + threadIdx.x;
    if (i < n) p[i] = 0.f;
}

__global__ void keys_init(const int* __restrict__ idx, unsigned* __restrict__ keys,
                          int* __restrict__ vals, int N)
{
    int i = blockIdx.x * blockDim.x + threadIdx.x;
    if (i < N) {
        const int* r = idx + (size_t)i * 4;
        keys[i] = (unsigned)(r[0] * STRIDE_B + r[1] * STRIDE_T + r[2] * STRIDE_H + r[3]);
        vals[i] = i;
    }
}

__global__ __launch_bounds__(256) void radix_hist(
    const unsigned* __restrict__ keys, int N, int shift,
    unsigned* __restrict__ gh, int NB)
{
    __shared__ unsigned cnt[16];
    if (threadIdx.x < 16) cnt[threadIdx.x] = 0;
    __syncthreads();
    int base = blockIdx.x * RELEMS + threadIdx.x * RITEMS;
    #pragma unroll
    for (int j = 0; j < RITEMS; ++j) {
        int i = base + j;
        if (i < N) atomicAdd(&cnt[(keys[i] >> shift) & 15], 1u);
    }
    __syncthreads();
    if (threadIdx.x < 16) gh[threadIdx.x * NB + blockIdx.x] = cnt[threadIdx.x];
}

__global__ __launch_bounds__(64) void radix_scan(unsigned* gh, int NB)
{
    __shared__ unsigned tot[16];
    __shared__ unsigned basev[16];
    int b = threadIdx.x;
    if (b < 16) {
        unsigned run = 0;
        for (int k = 0; k < NB; ++k) { unsigned v = gh[b * NB + k]; gh[b * NB + k] = run; run += v; }
        tot[b] = run;
    }
    __syncthreads();
    if (b == 0) { unsigned r = 0; for (int i = 0; i < 16; ++i) { basev[i] = r; r += tot[i]; } }
    __syncthreads();
    if (b < 16) { unsigned bb = basev[b]; for (int k = 0; k < NB; ++k) gh[b * NB + k] += bb; }
}

__global__ __launch_bounds__(256) void radix_scatter(
    const unsigned* __restrict__ keysIn, const int* __restrict__ valsIn,
    unsigned* __restrict__ keysOut, int* __restrict__ valsOut,
    int N, int shift, const unsigned* __restrict__ gh, int NB)
{
    __shared__ unsigned tc[256][16];
    __shared__ unsigned goff[16];
    int t = threadIdx.x;
    if (t < 16) goff[t] = gh[t * NB + blockIdx.x];

    int base = blockIdx.x * RELEMS + t * RITEMS;
    unsigned kk[RITEMS];
    unsigned myc[16];
    #pragma unroll
    for (int b = 0; b < 16; ++b) myc[b] = 0;
    #pragma unroll
    for (int j = 0; j < RITEMS; ++j) {
        int i = base + j;
        if (i < N) { unsigned K = keysIn[i]; kk[j] = K; myc[(K >> shift) & 15]++; }
        else kk[j] = 0xFFFFFFFFu;
    }
    #pragma unroll
    for (int b = 0; b < 16; ++b) tc[t][b] = myc[b];
    __syncthreads();
    if (t < 16) {
        unsigned run = 0;
        for (int t2 = 0; t2 < 256; ++t2) { unsigned v = tc[t2][t]; tc[t2][t] = run; run += v; }
    }
    __syncthreads();
    unsigned seq[16];
    #pragma unroll
    for (int b = 0; b < 16; ++b) seq[b] = 0;
    #pragma unroll
    for (int j = 0; j < RITEMS; ++j) {
        int i = base + j;
        if (i < N) {
            unsigned K = kk[j];
            int d = (K >> shift) & 15;
            unsigned dst = goff[d] + tc[t][d] + seq[d];
            seq[d]++;
            keysOut[dst] = K;
            valsOut[dst] = valsIn[i];
        }
    }
}

__global__ __launch_bounds__(256) void head_kernel(
    const unsigned* __restrict__ keys, int* __restrict__ head,
    int* __restrict__ bsums, int N)
{
    __shared__ int sh[256];
    int i = blockIdx.x * 256 + threadIdx.x;
    int h = 0;
    if (i < N) h = (i == 0) || (keys[i] != keys[i - 1]);
    if (i < N) head[i] = h;
    sh[threadIdx.x] = h;
    for (int off = 128; off > 0; off >>= 1) {
        __syncthreads();
        if (threadIdx.x < off) sh[threadIdx.x] += sh[threadIdx.x + off];
    }
    __syncthreads();
    if (threadIdx.x == 0) bsums[blockIdx.x] = sh[0];
}

__global__ void scan_bsums(int* bsums, int NBu, int* Utot)
{
    if (threadIdx.x == 0 && blockIdx.x == 0) {
        int r = 0;
        for (int k = 0; k < NBu; ++k) { int v = bsums[k]; bsums[k] = r; r += v; }
        *Utot = r;
    }
}

__global__ __launch_bounds__(256) void uid_kernel(
    const int* __restrict__ head, const int* __restrict__ bsums,
    int* __restrict__ uid, int N)
{
    __shared__ int sh[256];
    int i = blockIdx.x * 256 + threadIdx.x;
    sh[threadIdx.x] = (i < N) ? head[i] : 0;
    __syncthreads();
    for (int off = 1; off < 256; off <<= 1) {
        int v = 0;
        if (threadIdx.x >= off) v = sh[threadIdx.x - off];
        __syncthreads();
        sh[threadIdx.x] += v;
        __syncthreads();
    }
    if (i < N) uid[i] = bsums[blockIdx.x] + sh[threadIdx.x] - 1;
}

__global__ void fill_merged(float4* p, int total4) {
    int i = blockIdx.x * blockDim.x + threadIdx.x;
    if (i < total4) p[i] = make_float4(0.f, 0.f, 0.f, 0.f);
}

__global__ void fill_meta(int* comb, float* valid, int M) {
    int u = blockIdx.x * blockDim.x + threadIdx.x;
    if (u < M) {
        comb[u * 4 + 0] = 8;   // SENTINEL // STRIDE_B
        comb[u * 4 + 1] = 0;
        comb[u * 4 + 2] = 0;
        comb[u * 4 + 3] = 0;
        valid[u] = 0.f;
    }
}

// One wave per sorted position; heads gather their duplicate run.
__global__ __launch_bounds__(256) void finalize(
    const unsigned* __restrict__ keys, const int* __restrict__ vals,
    const int* __restrict__ head, const int* __restrict__ uid,
    const float* __restrict__ X2, const float* __restrict__ feats,
    const float* __restrict__ scale,
    float* __restrict__ merged, int* __restrict__ comb, float* __restrict__ valid, int N)
{
    int p    = (blockIdx.x * blockDim.x + threadIdx.x) >> 5;
    int lane = threadIdx.x & 31;
    if (p >= N) return;
    if (!head[p]) return;

    unsigned K = keys[p];
    int u = uid[p];
    int c = lane * 4;
    float4 sc = *(const float4*)(scale + c);
    float4 a = make_float4(0.f, 0.f, 0.f, 0.f);
    int j = p;
    while (j < N && keys[j] == K) {
        int i = vals[j];
        float4 o = *(const float4*)(X2 + (size_t)i * C + c);
        float4 f = *(const float4*)(feats + (size_t)i * C + c);
        a.x += o.x * sc.x + f.x;
        a.y += o.y * sc.y + f.y;
        a.z += o.z * sc.z + f.z;
        a.w += o.w * sc.w + f.w;
        ++j;
    }
    float s = fabsf(a.x) + fabsf(a.y) + fabsf(a.z) + fabsf(a.w);
    #pragma unroll
    for (int off = 1; off < 32; off <<= 1) s += __shfl_xor(s, off, 32);
    bool vld = (s > 1e-8f);

    float4 r;
    r.x = vld ? fmaxf(a.x, 0.f) : 0.f;
    r.y = vld ? fmaxf(a.y, 0.f) : 0.f;
    r.z = vld ? fmaxf(a.z, 0.f) : 0.f;
    r.w = vld ? fmaxf(a.w, 0.f) : 0.f;
    *(float4*)(merged + (size_t)u * C + c) = r;

    if (lane == 0) {
        comb[u * 4 + 0] = (int)(K >> 22);
        comb[u * 4 + 1] = (int)((K >> 18) & 15u);
        comb[u * 4 + 2] = (int)((K >> 9) & 511u);
        comb[u * 4 + 3] = (int)(K & 511u);
        valid[u] = vld ? 1.f : 0.f;
    }
}

// ---------------------------------------------------------------------------
extern "C" void kernel_launch(void* const* d_in, const int* in_sizes, int n_in,
                              void* d_out, int out_size, void* d_ws, size_t ws_size,
                              hipStream_t stream)
{
    const float* feats = (const float*)d_in[0];
    const int*   indices = (const int*)d_in[1];
    const float* W1   = (const float*)d_in[2];
    const float* ln1g = (const float*)d_in[3];
    const float* ln1b = (const float*)d_in[4];
    const float* W2   = (const float*)d_in[5];
    const float* ln2g = (const float*)d_in[6];
    const float* ln2b = (const float*)d_in[7];
    const float* sew1 = (const float*)d_in[8];
    const float* seb1 = (const float*)d_in[9];
    const float* sew2 = (const float*)d_in[10];
    const float* seb2 = (const float*)d_in[11];

    const int N = in_sizes[0] / C;        // 300000
    const int M = 2 * N;                  // padded output rows

    // workspace carve-up
    size_t off = 0;
    auto alloc = [&](size_t bytes) -> char* {
        size_t o = (off + 255) & ~(size_t)255;
        off = o + bytes;
        return (char*)d_ws + o;
    };
    float*    X2     = (float*)   alloc((size_t)N * C * sizeof(float));
    float*    W1p    = (float*)   alloc((size_t)C * C * sizeof(float));
    float*    W2p    = (float*)   alloc((size_t)C * C * sizeof(float));
    float*    colsum = (float*)   alloc(C * sizeof(float));
    float*    scale  = (float*)   alloc(C * sizeof(float));
    unsigned* keysA  = (unsigned*)alloc((size_t)N * sizeof(unsigned));
    int*      valsA  = (int*)     alloc((size_t)N * sizeof(int));
    unsigned* keysB  = (unsigned*)alloc((size_t)N * sizeof(unsigned));
    int*      valsB  = (int*)     alloc((size_t)N * sizeof(int));
    const int NB     = (N + RELEMS - 1) / RELEMS;
    unsigned* gh     = (unsigned*)alloc((size_t)16 * NB * sizeof(unsigned));
    int*      head   = (int*)     alloc((size_t)N * sizeof(int));
    int*      uid    = (int*)     alloc((size_t)N * sizeof(int));
    const int NBu    = (N + 255) / 256;
    int*      bsums  = (int*)     alloc((size_t)NBu * sizeof(int));
    int*      Utot   = (int*)     alloc(sizeof(int));
    (void)ws_size;

    // output carve-up: merged | comb_idx (int) | valid (float)
    float* merged = (float*)d_out;
    int*   comb   = (int*)((float*)d_out + (size_t)M * C);
    float* valid  = (float*)d_out + (size_t)M * C + (size_t)M * 4;

    // 1) zero colsum, pack weights into fragment order
    zero_f<<<1, 128, 0, stream>>>(colsum, C);
    pack_w<<<(32 * 8 * 32 + 255) / 256, 256, 0, stream>>>(W1, W1p);
    pack_w<<<(32 * 8 * 32 + 255) / 256, 256, 0, stream>>>(W2, W2p);

    // 2) fused MLP (both GEMMs + LNs, WMMA f32 16x16x4) + SE column sums
    {
        int tiles = (N + 15) / 16;
        int blocks = (tiles + 1) / 2;
        fused_mlp<<<blocks, 64, 0, stream>>>(feats, W1p, ln1g, ln1b, W2p, ln2g, ln2b,
                                             X2, colsum, N);
    }

    // 3) SE gate
    se_kernel<<<1, 128, 0, stream>>>(colsum, sew1, seb1, sew2, seb2, scale, N);

    // 4) build keys, radix sort (7 x 4-bit LSD, keys < 2^25)
    keys_init<<<(N + 255) / 256, 256, 0, stream>>>(indices, keysA, valsA, N);
    unsigned* ki = keysA; int* vi = valsA;
    unsigned* ko = keysB; int* vo = valsB;
    for (int pass = 0; pass < 7; ++pass) {
        int shift = pass * 4;
        radix_hist<<<NB, 256, 0, stream>>>(ki, N, shift, gh, NB);
        radix_scan<<<1, 64, 0, stream>>>(gh, NB);
        radix_scatter<<<NB, 256, 0, stream>>>(ki, vi, ko, vo, N, shift, gh, NB);
        unsigned* tk = ki; ki = ko; ko = tk;
        int* tv = vi; vi = vo; vo = tv;
    }
    // sorted result now in (ki, vi)

    // 5) unique ids
    head_kernel<<<NBu, 256, 0, stream>>>(ki, head, bsums, N);
    scan_bsums<<<1, 32, 0, stream>>>(bsums, NBu, Utot);
    uid_kernel<<<NBu, 256, 0, stream>>>(head, bsums, uid, N);

    // 6) default-fill padded outputs
    {
        int total4 = M * (C / 4);
        fill_merged<<<(total4 + 255) / 256, 256, 0, stream>>>((float4*)merged, total4);
        fill_meta<<<(M + 255) / 256, 256, 0, stream>>>(comb, valid, M);
    }

    // 7) gather duplicate runs, apply SE scale, ReLU, write outputs
    {
        long threads = (long)N * 32;
        int blocks = (int)((threads + 255) / 256);
        finalize<<<blocks, 256, 0, stream>>>(ki, vi, head, uid, X2, feats, scale,
                                             merged, comb, valid, N);
    }
}